// SparseConvEncoder_37761352466589
// MI455X (gfx1250) — compile-verified
//
#include <hip/hip_runtime.h>

// Sparse conv encoder for MI455X (gfx1250, wave32).
// FP32 matrix path via V_WMMA_F32_16X16X4_F32; gather staged in LDS (float4),
// each wave computes a 16x32 output tile (2 accumulators sharing the A
// fragment); scatter via global f32 atomics (indices have duplicates).

#define KOFF 27

typedef __attribute__((ext_vector_type(2))) float v2f;
typedef __attribute__((ext_vector_type(4))) float v4f;
typedef __attribute__((ext_vector_type(8))) float v8f;

// ---------------------------------------------------------------- zero fill
__global__ __launch_bounds__(256) void fill_zero_kernel(float* __restrict__ p, int n) {
  int i = blockIdx.x * blockDim.x + threadIdx.x;
  if (i < n) p[i] = 0.0f;
}

// ---------------------------------------------------------- sparse conv GEMM
// grid = ( ceil(M/16), 27, Cout/64 ), block = 64 (2 waves)
// Wave w handles columns [z*64 + w*32, +32): two 16x16 wmma accumulators
// sharing one A fragment, reduced over Cin in K=4 steps.
// dynamic LDS: 128B header (gather/scatter row ids) + 16 x (Cin+4) floats.
__global__ __launch_bounds__(64) void sparse_conv_wmma_kernel(
    const float* __restrict__ feats,  // [N_in, Cin]
    const float* __restrict__ W,      // [K, Cin, Cout]
    const int*   __restrict__ gidx,   // [K, M]
    const int*   __restrict__ sidx,   // [K, M]
    float*       __restrict__ out,    // [M, Cout] pre-zeroed accumulator
    int M, int Cin, int Cout)
{
  extern __shared__ char smem[];
  int*   sg   = (int*)smem;        // 16 gather rows
  int*   ss   = sg + 16;           // 16 scatter rows
  float* tile = (float*)(smem + 128);
  const int lstride = Cin + 4;     // pad: stride mod 64 == 4 -> conflict-free

  const int k    = blockIdx.y;     // kernel offset 0..26
  const int mt   = blockIdx.x;     // 16-row tile of M
  const int tid  = threadIdx.x;
  const int wave = tid >> 5;       // 0..1
  const int lane = tid & 31;

  if (tid < 16) {
    int m = mt * 16 + tid;
    sg[tid] = (m < M) ? gidx[k * M + m] : 0;
  } else if (tid < 32) {
    int t = tid - 16;
    int m = mt * 16 + t;
    ss[t] = (m < M) ? sidx[k * M + m] : 0;
  }
  __syncthreads();

  // stage the 16 gathered rows into LDS as float4 (coalesced per row)
  const int c4n = Cin >> 2;
  for (int r = wave; r < 16; r += 2) {
    const v4f* src = (const v4f*)(feats + (size_t)sg[r] * Cin);
    v4f* dst = (v4f*)(tile + r * lstride);
    for (int c4 = lane; c4 < c4n; c4 += 32)
      dst[c4] = src[c4];
  }
  __syncthreads();

  const int n0  = blockIdx.z * 64 + wave * 32;   // 32-col strip of Cout
  const int row = lane & 15;
  const int kk  = (lane >> 4) * 2;  // K split: lanes 0-15 -> K 0/1, 16-31 -> K 2/3
  const float* arow = tile + row * lstride + kk;
  // B base pointers for the two 16-col subtiles, rows kk and kk+1
  const float* wb = W + (size_t)k * Cin * Cout + n0 + row;
  const float* w0 = wb + (size_t)kk * Cout;        // row kk
  const float* w1 = wb + (size_t)(kk + 1) * Cout;  // row kk+1
  const size_t step = (size_t)4 * Cout;            // advance 4 K-rows

  v8f acc0 = {};
  v8f acc1 = {};
#pragma unroll 4
  for (int c0 = 0; c0 < Cin; c0 += 4) {
    v2f a, b0, b1;
    a.x = arow[c0];
    a.y = arow[c0 + 1];
    b0.x = w0[0];  b0.y = w1[0];    // cols n0..n0+15
    b1.x = w0[16]; b1.y = w1[16];   // cols n0+16..n0+31
    w0 += step;
    w1 += step;
    // 8 args: (neg_a, A, neg_b, B, c_mod, C, reuse_a, reuse_b)
    acc0 = __builtin_amdgcn_wmma_f32_16x16x4_f32(
        false, a, false, b0, (short)0, acc0, false, false);
    acc1 = __builtin_amdgcn_wmma_f32_16x16x4_f32(
        false, a, false, b1, (short)0, acc1, false, false);
  }

  // C/D layout: vgpr v -> M=v (lanes 0-15) / M=v+8 (lanes 16-31); col = lane&15
  const int crowbase = (lane >> 4) * 8;
  const int col = n0 + (lane & 15);
#pragma unroll
  for (int v = 0; v < 8; ++v) {
    int cm = crowbase + v;                 // tile-local output row
    if (mt * 16 + cm < M) {
      float* orow = out + (size_t)ss[cm] * Cout;
      atomicAdd(orow + col,      acc0[v]);
      atomicAdd(orow + col + 16, acc1[v]);
    }
  }
}

// ------------------------------------------------- bias + LayerNorm (+res) + ReLU
// one wave per row; block = 256 -> 8 rows/block
__global__ __launch_bounds__(256) void bias_ln_relu_kernel(
    const float* __restrict__ in, const float* __restrict__ bias,
    const float* __restrict__ gamma, const float* __restrict__ beta,
    const float* __restrict__ residual,   // may be null
    float* __restrict__ out, int M, int C)
{
  int row = blockIdx.x * 8 + (threadIdx.x >> 5);
  if (row >= M) return;
  int lane = threadIdx.x & 31;
  const float* x = in + (size_t)row * C;

  float s = 0.0f, ss = 0.0f;
  for (int c = lane; c < C; c += 32) {
    float v = x[c] + bias[c];
    s += v; ss += v * v;
  }
#pragma unroll
  for (int off = 16; off > 0; off >>= 1) {
    s  += __shfl_xor(s,  off, 32);
    ss += __shfl_xor(ss, off, 32);
  }
  float inv  = 1.0f / (float)C;
  float mean = s * inv;
  float var  = ss * inv - mean * mean;
  float r    = rsqrtf(var + 1e-5f);

  float* o = out + (size_t)row * C;
  const float* res = residual ? residual + (size_t)row * C : nullptr;
  for (int c = lane; c < C; c += 32) {
    float v = (x[c] + bias[c] - mean) * r * gamma[c] + beta[c];
    if (res) v += res[c];
    o[c] = fmaxf(v, 0.0f);
  }
}

// ------------------------------------------------------------------ launcher
static inline void zero_buf(float* p, int n, hipStream_t s) {
  fill_zero_kernel<<<(n + 255) / 256, 256, 0, s>>>(p, n);
}

static inline void conv(const float* feats, const float* W,
                        const int* gidx, const int* sidx, float* out,
                        int M, int Cin, int Cout, hipStream_t s) {
  dim3 grid((M + 15) / 16, KOFF, Cout / 64);
  size_t shmem = 128 + (size_t)16 * (Cin + 4) * sizeof(float);
  sparse_conv_wmma_kernel<<<grid, 64, shmem, s>>>(feats, W, gidx, sidx, out, M, Cin, Cout);
}

static inline void ln(const float* in, const float* b, const float* g,
                      const float* bt, const float* res, float* out,
                      int M, int C, hipStream_t s) {
  bias_ln_relu_kernel<<<(M + 7) / 8, 256, 0, s>>>(in, b, g, bt, res, out, M, C);
}

extern "C" void kernel_launch(void* const* d_in, const int* in_sizes, int n_in,
                              void* d_out, int out_size, void* d_ws, size_t ws_size,
                              hipStream_t stream) {
  (void)in_sizes; (void)n_in; (void)out_size; (void)ws_size;

  static const int NS[5] = {80000, 40000, 20000, 10000, 5000};
  static const int CH[5] = {32, 64, 128, 192, 256};

  const float* feats = (const float*)d_in[0];

  // JAX pytree flatten order (sorted dict keys):
  // params.down[l]: W, b, ln_b, ln_g            -> indices 1 + 4l ...
  // params.res[l] : W1, W2, b1, b2, ln1_b, ln1_g, ln2_b, ln2_g -> 17 + 8l ...
  // then gidx_down{l}, sidx_down{l}, gidx_lvl{l}, sidx_lvl{l}  -> 49 + 4l ...
  const float *dW[4], *db[4], *dlnb[4], *dlng[4];
  const float *W1[4], *W2[4], *b1[4], *b2[4], *l1b[4], *l1g[4], *l2b[4], *l2g[4];
  const int *gd[4], *sd[4], *gl[4], *sl[4];
  for (int l = 0; l < 4; ++l) {
    dW[l]   = (const float*)d_in[1 + 4 * l + 0];
    db[l]   = (const float*)d_in[1 + 4 * l + 1];
    dlnb[l] = (const float*)d_in[1 + 4 * l + 2];
    dlng[l] = (const float*)d_in[1 + 4 * l + 3];
    W1[l]  = (const float*)d_in[17 + 8 * l + 0];
    W2[l]  = (const float*)d_in[17 + 8 * l + 1];
    b1[l]  = (const float*)d_in[17 + 8 * l + 2];
    b2[l]  = (const float*)d_in[17 + 8 * l + 3];
    l1b[l] = (const float*)d_in[17 + 8 * l + 4];
    l1g[l] = (const float*)d_in[17 + 8 * l + 5];
    l2b[l] = (const float*)d_in[17 + 8 * l + 6];
    l2g[l] = (const float*)d_in[17 + 8 * l + 7];
    gd[l] = (const int*)d_in[49 + 4 * l + 0];
    sd[l] = (const int*)d_in[49 + 4 * l + 1];
    gl[l] = (const int*)d_in[49 + 4 * l + 2];
    sl[l] = (const int*)d_in[49 + 4 * l + 3];
  }

  // output regions (flat concat of the returned tuple)
  float* out = (float*)d_out;
  size_t ooff[5];
  ooff[0] = 0;
  for (int l = 0; l < 4; ++l)
    ooff[l + 1] = ooff[l] + (size_t)NS[l] * CH[l];

  // workspace: bufA (max M*Cout = 2.56M floats), bufB (max M*Cmid = 5.12M floats)
  float* bufA = (float*)d_ws;
  float* bufB = bufA + 2560000;

  // output[0] = input feats, unchanged
  hipMemcpyAsync(out, feats, (size_t)NS[0] * CH[0] * sizeof(float),
                 hipMemcpyDeviceToDevice, stream);

  const float* xin = feats;  // level input features
  for (int l = 0; l < 4; ++l) {
    const int M = NS[l + 1], Cin = CH[l], Cout = CH[l + 1], Cmid = 2 * Cout;
    float* xlvl = out + ooff[l + 1];  // this level's output region (= residual)

    // down: conv -> bias+LN+ReLU
    zero_buf(bufA, M * Cout, stream);
    conv(xin, dW[l], gd[l], sd[l], bufA, M, Cin, Cout, stream);
    ln(bufA, db[l], dlng[l], dlnb[l], nullptr, xlvl, M, Cout, stream);

    // resblock conv1: Cout -> Cmid, bias+LN+ReLU (in place in bufB)
    zero_buf(bufB, M * Cmid, stream);
    conv(xlvl, W1[l], gl[l], sl[l], bufB, M, Cout, Cmid, stream);
    ln(bufB, b1[l], l1g[l], l1b[l], nullptr, bufB, M, Cmid, stream);

    // resblock conv2: Cmid -> Cout, bias+LN, +residual, ReLU -> xlvl
    zero_buf(bufA, M * Cout, stream);
    conv(bufB, W2[l], gl[l], sl[l], bufA, M, Cmid, Cout, stream);
    ln(bufA, b2[l], l2g[l], l2b[l], xlvl, xlvl, M, Cout, stream);

    xin = xlvl;
  }
}